// GraphConvoluationSparse_11235634446663
// MI455X (gfx1250) — compile-verified
//
#include <hip/hip_runtime.h>
#include <hip/hip_bf16.h>

typedef float v2f __attribute__((ext_vector_type(2)));
typedef float v8f __attribute__((ext_vector_type(8)));

#define DIM 256   // IN_DIM == OUT_DIM
#define CAP 96    // per-row edge bucket capacity (Poisson(32): P(deg>96) ~ 4e-20)

// ---------------------------------------------------------------------------
// Kernel 1: h = x @ w  via V_WMMA_F32_16X16X4_F32.
// Block = 256 threads = 8 waves; each wave owns a 16(M) x 64(N) slab.
// Row index is CLAMPED (not predicated): an out-of-range lane only feeds an
// out-of-range M row of the tile, which is never stored -> no EXEC juggling
// in the hot loop.
// ---------------------------------------------------------------------------
__global__ __launch_bounds__(256)
void gemm_xw(const float* __restrict__ x, const float* __restrict__ w,
             float* __restrict__ h, int n_nodes) {
    const int tid    = threadIdx.x;
    const int wave   = tid >> 5;
    const int lane   = tid & 31;
    const int lane16 = lane & 15;
    const int hi     = lane >> 4;             // 0: K pair {0,1}, 1: K pair {2,3}

    const int mtile    = blockIdx.x * 2 + (wave >> 2);
    const int n0       = (wave & 3) * 64;
    const int row_base = mtile * 16;

    int arow = row_base + lane16;
    if (arow >= n_nodes) arow = n_nodes - 1;          // clamp, not predicate
    const float* aptr = x + (size_t)arow * DIM + 2 * hi;

    v8f acc0 = {}; v8f acc1 = {}; v8f acc2 = {}; v8f acc3 = {};

    for (int k = 0; k < DIM; k += 4) {
        const v2f a = *(const v2f*)(aptr + k);        // x[row][k+2hi .. +1]

        const float* wrow0 = w + (size_t)(k + 2 * hi) * DIM;
        const float* wrow1 = wrow0 + DIM;

        v2f b0, b1, b2, b3;
        b0.x = wrow0[n0 +  0 + lane16]; b0.y = wrow1[n0 +  0 + lane16];
        b1.x = wrow0[n0 + 16 + lane16]; b1.y = wrow1[n0 + 16 + lane16];
        b2.x = wrow0[n0 + 32 + lane16]; b2.y = wrow1[n0 + 32 + lane16];
        b3.x = wrow0[n0 + 48 + lane16]; b3.y = wrow1[n0 + 48 + lane16];

        acc0 = __builtin_amdgcn_wmma_f32_16x16x4_f32(false, a, false, b0, (short)0, acc0, false, false);
        acc1 = __builtin_amdgcn_wmma_f32_16x16x4_f32(false, a, false, b1, (short)0, acc1, false, false);
        acc2 = __builtin_amdgcn_wmma_f32_16x16x4_f32(false, a, false, b2, (short)0, acc2, false, false);
        acc3 = __builtin_amdgcn_wmma_f32_16x16x4_f32(false, a, false, b3, (short)0, acc3, false, false);
    }

    #pragma unroll
    for (int r = 0; r < 8; ++r) {
        const int row = row_base + r + 8 * hi;
        if (row < n_nodes) {
            float* hrow = h + (size_t)row * DIM + n0 + lane16;
            hrow[ 0] = acc0[r];
            hrow[16] = acc1[r];
            hrow[32] = acc2[r];
            hrow[48] = acc3[r];
        }
    }
}

// ---------------------------------------------------------------------------
// Kernel 2: zero the per-row degree counters (re-done every call).
// ---------------------------------------------------------------------------
__global__ __launch_bounds__(256)
void zero_counts(int* __restrict__ counts, int n_nodes) {
    const int i = blockIdx.x * 256 + threadIdx.x;
    if (i < n_nodes) counts[i] = 0;
}

// ---------------------------------------------------------------------------
// Kernel 3: bin edges by destination row. Only INTEGER atomics (3.2M of them,
// native global_atomic_add_u32) instead of 819M fp32 atomics.
// ---------------------------------------------------------------------------
__global__ __launch_bounds__(256)
void bucket_build(const int* __restrict__ erow, int* __restrict__ counts,
                  int* __restrict__ pos, int* __restrict__ bucket, int n_edges) {
    const int e = blockIdx.x * 256 + threadIdx.x;
    if (e >= n_edges) return;
    const int r = erow[e];
    const int p = atomicAdd(&counts[r], 1);
    pos[e] = p;
    if (p < CAP) bucket[(size_t)r * CAP + p] = e;
}

// ---------------------------------------------------------------------------
// Kernel 4: pull-style row reduction. 64 threads (2 waves) per row, 4 rows per
// block. Edge (col, val) metadata is staged in LDS so the inner loop issues
// only the 16B h-row gathers (L2-resident, several in flight via unroll).
// out[row] = b + sum(val * h[col])  -- single store, zero fp atomics.
// ---------------------------------------------------------------------------
__global__ __launch_bounds__(256)
void row_reduce(const int* __restrict__ ecol, const float* __restrict__ evals,
                const float* __restrict__ h, const int* __restrict__ counts,
                const int* __restrict__ bucket, const float* __restrict__ b,
                float* __restrict__ out, int n_nodes) {
    __shared__ int   s_col[4][CAP];
    __shared__ float s_val[4][CAP];

    const int tid  = threadIdx.x;
    const int sub  = tid >> 6;         // row within block (0..3)
    const int lane = tid & 63;         // 64 lanes x float4 = 256 floats

    const int  row   = blockIdx.x * 4 + sub;
    const bool valid = row < n_nodes;
    const int  deg   = valid ? min(counts[row], CAP) : 0;

    const int* bk = bucket + (size_t)row * CAP;
    for (int i = lane; i < deg; i += 64) {
        const int e  = bk[i];
        s_col[sub][i] = ecol[e];
        s_val[sub][i] = evals[e];
    }
    __syncthreads();

    float4 acc = ((const float4*)b)[lane];

    int i = 0;
    for (; i + 4 <= deg; i += 4) {
        const int   c0 = s_col[sub][i + 0], c1 = s_col[sub][i + 1];
        const int   c2 = s_col[sub][i + 2], c3 = s_col[sub][i + 3];
        const float v0 = s_val[sub][i + 0], v1 = s_val[sub][i + 1];
        const float v2 = s_val[sub][i + 2], v3 = s_val[sub][i + 3];
        const float4 h0 = ((const float4*)(h + (size_t)c0 * DIM))[lane];
        const float4 h1 = ((const float4*)(h + (size_t)c1 * DIM))[lane];
        const float4 h2 = ((const float4*)(h + (size_t)c2 * DIM))[lane];
        const float4 h3 = ((const float4*)(h + (size_t)c3 * DIM))[lane];
        acc.x += v0 * h0.x; acc.y += v0 * h0.y; acc.z += v0 * h0.z; acc.w += v0 * h0.w;
        acc.x += v1 * h1.x; acc.y += v1 * h1.y; acc.z += v1 * h1.z; acc.w += v1 * h1.w;
        acc.x += v2 * h2.x; acc.y += v2 * h2.y; acc.z += v2 * h2.z; acc.w += v2 * h2.w;
        acc.x += v3 * h3.x; acc.y += v3 * h3.y; acc.z += v3 * h3.z; acc.w += v3 * h3.w;
    }
    for (; i < deg; ++i) {
        const int    c  = s_col[sub][i];
        const float  v  = s_val[sub][i];
        const float4 hv = ((const float4*)(h + (size_t)c * DIM))[lane];
        acc.x += v * hv.x; acc.y += v * hv.y; acc.z += v * hv.z; acc.w += v * hv.w;
    }

    if (valid) ((float4*)(out + (size_t)row * DIM))[lane] = acc;
}

// ---------------------------------------------------------------------------
// Kernel 5: handle (practically nonexistent) bucket overflow. One thread per
// edge checks its recorded slot; only overflowing edges do fp32 atomics.
// ---------------------------------------------------------------------------
__global__ __launch_bounds__(256)
void overflow_fix(const int* __restrict__ erow, const int* __restrict__ ecol,
                  const float* __restrict__ evals, const float* __restrict__ h,
                  const int* __restrict__ pos, float* __restrict__ out, int n_edges) {
    const int e = blockIdx.x * 256 + threadIdx.x;
    if (e >= n_edges) return;
    if (pos[e] < CAP) return;                       // common case: nothing to do
    const int   row = erow[e];
    const int   col = ecol[e];
    const float v   = evals[e];
    const float* hp = h + (size_t)col * DIM;
    float*       op = out + (size_t)row * DIM;
    for (int d = 0; d < DIM; ++d)
        __hip_atomic_fetch_add(op + d, v * hp[d], __ATOMIC_RELAXED, __HIP_MEMORY_SCOPE_AGENT);
}

// ---------------------------------------------------------------------------
// Host launcher.
// Inputs: x[N,256] f32, edge_row[E] i32, edge_col[E] i32, edge_vals[E] f32,
// w[256,256] f32, b[256] f32. Output: out[N,256] f32.
// Workspace layout: h | counts | pos | bucket  (~154 MB for N=1e5, E=3.2e6).
// ---------------------------------------------------------------------------
extern "C" void kernel_launch(void* const* d_in, const int* in_sizes, int n_in,
                              void* d_out, int out_size, void* d_ws, size_t ws_size,
                              hipStream_t stream) {
    const float* x     = (const float*)d_in[0];
    const int*   erow  = (const int*)  d_in[1];
    const int*   ecol  = (const int*)  d_in[2];
    const float* evals = (const float*)d_in[3];
    const float* w     = (const float*)d_in[4];
    const float* b     = (const float*)d_in[5];
    float*       out   = (float*)d_out;

    const int n_nodes = in_sizes[0] / DIM;
    const int n_edges = in_sizes[1];

    char* ws = (char*)d_ws;
    size_t off = 0;
    float* h      = (float*)(ws + off); off += ((size_t)n_nodes * DIM * 4 + 255) & ~(size_t)255;
    int*   counts = (int*)  (ws + off); off += ((size_t)n_nodes * 4 + 255) & ~(size_t)255;
    int*   pos    = (int*)  (ws + off); off += ((size_t)n_edges * 4 + 255) & ~(size_t)255;
    int*   bucket = (int*)  (ws + off);

    // 1) h = x @ w  (32 rows per block)
    gemm_xw<<<(n_nodes + 31) / 32, 256, 0, stream>>>(x, w, h, n_nodes);

    // 2) CSR-ish binning by destination row
    zero_counts<<<(n_nodes + 255) / 256, 256, 0, stream>>>(counts, n_nodes);
    bucket_build<<<(n_edges + 255) / 256, 256, 0, stream>>>(erow, counts, pos, bucket, n_edges);

    // 3) pull-style reduction: out[row] = b + sum val*h[col]
    row_reduce<<<(n_nodes + 3) / 4, 256, 0, stream>>>(ecol, evals, h, counts, bucket, b, out, n_nodes);

    // 4) overflow fallback (no-op in practice)
    overflow_fix<<<(n_edges + 255) / 256, 256, 0, stream>>>(erow, ecol, evals, h, pos, out, n_edges);
}